// MaskLM_49538152792852
// MI455X (gfx1250) — compile-verified
//
#include <hip/hip_runtime.h>

// ---------------------------------------------------------------------------
// MaskLM head for MI455X (gfx1250, wave32).
// Roofline: ~470MB traffic (W2 131MB read + out 327MB write) @ 23.3TB/s ≈ 20us;
// only ~9 TFLOPS of the 174 GFLOP total needed -> memory bound. So: keep full
// fp32 precision via V_WMMA_F32_16X16X4_F32, and move tiles with the Tensor
// Data Mover (async DMA -> LDS, TENSORcnt-fenced) so the VMEM pipe is free
// for the output stream and the main loop is pure ds_load + v_wmma.
// ---------------------------------------------------------------------------

typedef __attribute__((ext_vector_type(2))) float v2f;
typedef __attribute__((ext_vector_type(4))) float f32x4;
typedef __attribute__((ext_vector_type(8))) float v8f;
typedef __attribute__((ext_vector_type(4))) unsigned int u32x4;
typedef __attribute__((ext_vector_type(4))) int i32x4;
typedef __attribute__((ext_vector_type(8))) int i32x8;

#define B_ 8
#define S_ 2048
#define H_ 1024
#define P_ 320
#define V_ 32000
#define M_ (B_ * P_)   // 2560 gathered rows

// ---------------------------------------------------------------------------
// Tensor Data Mover: build a 2D-tile D# (ISA ch.8) and issue TENSOR_LOAD_TO_LDS.
//   group0: count=1 | lds_addr | global_addr[56:0] | type=2
//   group1: data_size=4B, pad_enable + pad codes, tensor dims, tile dims,
//           row stride (elements)
// pad_interval_code: row length = 2^(code+1) dwords ; pad_amount_code: (code+1) dwords
// ---------------------------------------------------------------------------
__device__ __forceinline__ void tdm_load_2d_f32(
    unsigned lds_off, const float* gptr,
    unsigned tile_w, unsigned tile_h,
    unsigned tensor_w, unsigned tensor_h,
    unsigned row_stride_elems,
    unsigned pad_interval_code, unsigned pad_amount_code)
{
    unsigned long long ga = (unsigned long long)(uintptr_t)gptr;
    u32x4 g0;
    g0.x = 1u;                                                  // count=1 (valid)
    g0.y = lds_off;                                             // lds_addr (bytes)
    g0.z = (unsigned)(ga & 0xFFFFFFFFull);                      // global_addr[31:0]
    g0.w = (unsigned)((ga >> 32) & 0x01FFFFFFull) | (2u << 30); // addr[56:32] | type=2
    i32x8 g1;
    g1[0] = (int)((2u << 16)                    // data_size = 4 bytes
                  | (1u << 20)                  // pad_enable (write into padded LDS rows)
                  | (pad_interval_code << 22)
                  | (pad_amount_code << 25));
    g1[1] = (int)((tensor_w & 0xFFFFu) << 16);                          // dim0[15:0]
    g1[2] = (int)(((tensor_w >> 16) & 0xFFFFu) | ((tensor_h & 0xFFFFu) << 16));
    g1[3] = (int)(((tensor_h >> 16) & 0xFFFFu) | ((tile_w & 0xFFFFu) << 16));
    g1[4] = (int)(tile_h & 0xFFFFu);                                    // tile_dim1
    g1[5] = (int)row_stride_elems;                                      // dim0_stride lo
    g1[6] = 0;                                                          // stride hi / dim1_stride
    g1[7] = 0;
    i32x4 gz = {0, 0, 0, 0};
#if defined(__clang_major__) && (__clang_major__ >= 23)
    i32x8 gz8 = {0, 0, 0, 0, 0, 0, 0, 0};
    __builtin_amdgcn_tensor_load_to_lds(g0, g1, gz, gz, gz8, 0);
#else
    __builtin_amdgcn_tensor_load_to_lds(g0, g1, gz, gz, 0);
#endif
}

// ---------------------------------------------------------------------------
// Kernel 1: gather + GEMM1 (16x1024 @ 1024x1024) + bias + ReLU + LayerNorm.
// One workgroup = 16 output rows, all 1024 cols. 8 waves, 128 cols per wave.
// ---------------------------------------------------------------------------
#define LNPAD 4
__global__ __launch_bounds__(256) void mlm_gemm1_ln(
    const float* __restrict__ X, const int* __restrict__ pos,
    const float* __restrict__ W1, const float* __restrict__ b1,
    const float* __restrict__ gamma, const float* __restrict__ beta,
    float* __restrict__ Hout)
{
    __shared__ float sA[16][H_ + LNPAD];

    const int tid  = threadIdx.x;
    const int row0 = blockIdx.x * 16;

    // Gather 16 rows X[b, pos[b,p], :] into LDS (float4, fully coalesced).
    for (int i = tid; i < 16 * (H_ / 4); i += 256) {
        int r  = i >> 8;
        int c4 = i & 255;
        int m  = row0 + r;
        int b  = m / P_;
        int p  = m - b * P_;
        int s  = pos[b * P_ + p];
        const f32x4* src = (const f32x4*)(X + ((size_t)b * S_ + (size_t)s) * H_);
        *(f32x4*)&sA[r][c4 * 4] = src[c4];
    }
    __syncthreads();

    const int wave = tid >> 5;
    const int lane = tid & 31;
    const int half = lane >> 4;          // lanes 16-31: K+2..K+3 / M+8..M+15
    const int ln16 = lane & 15;
    const int col0 = wave * 128;

    v8f acc[8] = {};                     // 16 x 128 output slab per wave

    for (int k = 0; k < H_; k += 4) {
        v2f afrag = *(const v2f*)&sA[ln16][k + 2 * half];
        const float* bcol = W1 + (size_t)(k + 2 * half) * H_ + col0 + ln16;
#pragma unroll
        for (int t = 0; t < 8; ++t) {
            v2f bfrag;
            bfrag.x = bcol[t * 16];
            bfrag.y = bcol[t * 16 + H_];
            acc[t] = __builtin_amdgcn_wmma_f32_16x16x4_f32(
                false, afrag, false, bfrag, (short)0, acc[t], false, false);
        }
    }
    __syncthreads();   // done reading A tile; reuse LDS for h

    // bias + ReLU -> LDS (C layout: VGPR r <-> M = r + 8*half, N = ln16)
#pragma unroll
    for (int t = 0; t < 8; ++t) {
        int col    = col0 + t * 16 + ln16;
        float bias = b1[col];
#pragma unroll
        for (int r = 0; r < 8; ++r) {
            float v = acc[t][r] + bias;
            sA[8 * half + r][col] = v > 0.f ? v : 0.f;
        }
    }
    __syncthreads();

    // LayerNorm: 2 rows per wave, lane-parallel + shuffle reduction (wave32).
#pragma unroll
    for (int rr = 0; rr < 2; ++rr) {
        int r = wave * 2 + rr;
        float sum = 0.f, sq = 0.f;
        for (int c = lane; c < H_; c += 32) {
            float v = sA[r][c];
            sum += v; sq += v * v;
        }
#pragma unroll
        for (int off = 16; off > 0; off >>= 1) {
            sum += __shfl_xor(sum, off, 32);
            sq  += __shfl_xor(sq,  off, 32);
        }
        float mu  = sum * (1.0f / H_);
        float var = sq * (1.0f / H_) - mu * mu;
        float rs  = rsqrtf(var + 1e-5f);
        float* dst = Hout + (size_t)(row0 + r) * H_;
        for (int c = lane; c < H_; c += 32) {
            dst[c] = (sA[r][c] - mu) * rs * gamma[c] + beta[c];
        }
    }
}

// ---------------------------------------------------------------------------
// Kernel 2: GEMM2  h[2560x1024] @ W2[1024x32000] + b2 -> out[2560x32000].
// 64x128 tile per workgroup, BK=32, double-buffered LDS fed by the TDM:
//   wave 0 issues the B tile (32x128 of W2), wave 1 the A tile (64x32 of h).
// Main loop is pure ds_load + v_wmma; TENSORcnt fences the buffers.
// ---------------------------------------------------------------------------
#define BM 64
#define BN 128
#define BK 32
#define PADA 4   // A row = 32 dwords + 4 pad  -> interval code 4, amount code 3
#define PADB 4   // B row = 128 dwords + 4 pad -> interval code 6, amount code 3

__global__ __launch_bounds__(256) void mlm_gemm2(
    const float* __restrict__ Hin, const float* __restrict__ W2,
    const float* __restrict__ b2, float* __restrict__ out)
{
    __shared__ float sA[2][BM][BK + PADA];
    __shared__ float sB[2][BK][BN + PADB];

    const int tid  = threadIdx.x;
    const int n0   = blockIdx.x * BN;
    const int m0   = blockIdx.y * BM;
    const int wave = tid >> 5;
    const int lane = tid & 31;
    const int half = lane >> 4;
    const int ln16 = lane & 15;
    const int wm   = wave >> 1;          // 0..3 -> row block
    const int wn   = wave & 1;           // 0..1 -> col block

    auto issueTiles = [&](int kt, int buf) {
        const int k0 = kt * BK;
        if (wave == 0) {
            // B tile: 32 rows x 128 cols of W2, row stride V_
            tdm_load_2d_f32((unsigned)(uintptr_t)&sB[buf][0][0],
                            W2 + (size_t)k0 * V_ + n0,
                            /*tile_w=*/BN, /*tile_h=*/BK,
                            /*tensor_w=*/V_, /*tensor_h=*/H_,
                            /*row_stride=*/V_,
                            /*pad_interval=*/6, /*pad_amount=*/3);
        } else if (wave == 1) {
            // A tile: 64 rows x 32 k of h, row stride H_
            tdm_load_2d_f32((unsigned)(uintptr_t)&sA[buf][0][0],
                            Hin + (size_t)m0 * H_ + k0,
                            /*tile_w=*/BK, /*tile_h=*/BM,
                            /*tensor_w=*/H_, /*tensor_h=*/M_,
                            /*row_stride=*/H_,
                            /*pad_interval=*/4, /*pad_amount=*/3);
        }
    };

    v8f acc[4] = {};

    issueTiles(0, 0);
    __builtin_amdgcn_s_wait_tensorcnt(0);   // no-op for non-producer waves
    __syncthreads();

    const int NKT = H_ / BK;             // 32 K-chunks
    for (int kt = 0; kt < NKT; ++kt) {
        const int cur = kt & 1;
        if (kt + 1 < NKT) issueTiles(kt + 1, cur ^ 1);   // DMA next tiles during compute
#pragma unroll
        for (int kk = 0; kk < BK; kk += 4) {
            const int ks = kk + 2 * half;
            v2f afrag = *(const v2f*)&sA[cur][wm * 16 + ln16][ks];
#pragma unroll
            for (int t = 0; t < 4; ++t) {
                int nn = wn * 64 + t * 16 + ln16;
                v2f bfrag;
                bfrag.x = sB[cur][ks][nn];
                bfrag.y = sB[cur][ks + 1][nn];
                acc[t] = __builtin_amdgcn_wmma_f32_16x16x4_f32(
                    false, afrag, false, bfrag, (short)0, acc[t], false, false);
            }
        }
        __builtin_amdgcn_s_wait_tensorcnt(0);   // producer waves fence their DMA
        __syncthreads();
    }

    // Epilogue: + b2, store (lanes 0-15 hit contiguous columns)
#pragma unroll
    for (int t = 0; t < 4; ++t) {
        int col    = n0 + wn * 64 + t * 16 + ln16;
        float bias = b2[col];
        float* dst = out + (size_t)(m0 + wm * 16 + 8 * half) * V_ + col;
#pragma unroll
        for (int r = 0; r < 8; ++r) {
            dst[(size_t)r * V_] = acc[t][r] + bias;
        }
    }
}

// ---------------------------------------------------------------------------
extern "C" void kernel_launch(void* const* d_in, const int* in_sizes, int n_in,
                              void* d_out, int out_size, void* d_ws, size_t ws_size,
                              hipStream_t stream) {
    (void)in_sizes; (void)n_in; (void)out_size; (void)ws_size;
    const float* X     = (const float*)d_in[0];
    const int*   pos   = (const int*)d_in[1];
    const float* W1    = (const float*)d_in[2];
    const float* b1    = (const float*)d_in[3];
    const float* gam   = (const float*)d_in[4];
    const float* bet   = (const float*)d_in[5];
    const float* W2    = (const float*)d_in[6];
    const float* b2    = (const float*)d_in[7];
    float*       out   = (float*)d_out;
    float*       Hws   = (float*)d_ws;          // 2560 x 1024 fp32 = 10.5 MB

    mlm_gemm1_ln<<<M_ / 16, 256, 0, stream>>>(X, pos, W1, b1, gam, bet, Hws);
    mlm_gemm2<<<dim3(V_ / BN, M_ / BM), 256, 0, stream>>>(Hws, W2, b2, out);
}